// BatchedCauchyKernel_CONCERT_flex_3891240370725
// MI455X (gfx1250) — compile-verified
//
#include <hip/hip_runtime.h>
#include <hip/hip_bf16.h>
#include <math.h>

typedef __attribute__((ext_vector_type(2))) float v2f;
typedef __attribute__((ext_vector_type(4))) float v4f;
typedef __attribute__((ext_vector_type(8))) float v8f;

#define NROWS 4096
#define DDIM  128
#define SDIM  16
#define TILE_M 128
#define TILE_N 128
#define KC 32
#define LDSS 36   // padded row stride in floats: 36*4=144B (16B aligned, bank-conflict-free)

// ---------------------------------------------------------------------------
// Phase 1: scale_x = clip(sample @ clip(scale), 1e-6); xs = x * rsqrt(scale_x);
//          x2[row] = sum(xs^2). One block (128 threads) per row.
// ---------------------------------------------------------------------------
__global__ __launch_bounds__(128) void scale_rows_kernel(
    const float* __restrict__ src,    // [4096,128]  x or y
    const float* __restrict__ samp,   // [4096,16]
    const float* __restrict__ scale,  // [16,128]
    float* __restrict__ out_scaled,   // [4096,128]
    float* __restrict__ out_sq)       // [4096]
{
    const int row = blockIdx.x;
    const int d   = threadIdx.x;       // 0..127

    float acc = 0.f;
#pragma unroll
    for (int s = 0; s < SDIM; ++s) {
        float sc = fminf(fmaxf(scale[s * DDIM + d], 1e-6f), 1e6f);
        acc = fmaf(samp[row * SDIM + s], sc, acc);
    }
    acc = fmaxf(acc, 1e-6f);
    float v = src[row * DDIM + d] * rsqrtf(acc);
    out_scaled[row * DDIM + d] = v;

    float sq = v * v;
#pragma unroll
    for (int off = 16; off > 0; off >>= 1) sq += __shfl_xor(sq, off, 32);
    __shared__ float part[4];
    if ((threadIdx.x & 31) == 0) part[threadIdx.x >> 5] = sq;
    __syncthreads();
    if (threadIdx.x == 0) out_sq[row] = part[0] + part[1] + part[2] + part[3];
}

// ---------------------------------------------------------------------------
// mean(cutoff) -> workspace scalar (for the where(mean>0, ...) gate)
// ---------------------------------------------------------------------------
__global__ __launch_bounds__(256) void mean_kernel(
    const float* __restrict__ cutoff, float* __restrict__ out, int n)
{
    float s = 0.f;
    for (int i = threadIdx.x; i < n; i += 256) s += cutoff[i];
#pragma unroll
    for (int off = 16; off > 0; off >>= 1) s += __shfl_xor(s, off, 32);
    __shared__ float part[8];
    if ((threadIdx.x & 31) == 0) part[threadIdx.x >> 5] = s;
    __syncthreads();
    if (threadIdx.x == 0) {
        float t = 0.f;
#pragma unroll
        for (int i = 0; i < 8; ++i) t += part[i];
        *out = t / (float)n;
    }
}

// ---------------------------------------------------------------------------
// Main: xy = xs @ ys^T via V_WMMA_F32_16X16X4_F32, fused Cauchy + cutoff epilogue.
// 256 threads = 8 waves; block tile 128x128; wave tile 32x64 (2x4 of 16x16).
// ---------------------------------------------------------------------------
__global__ __launch_bounds__(256) void cauchy_gemm_kernel(
    const float* __restrict__ xs, const float* __restrict__ ys,
    const float* __restrict__ x2, const float* __restrict__ y2,
    const float* __restrict__ cutoff, const float* __restrict__ cmean,
    float* __restrict__ out)
{
    __shared__ float As[TILE_M * LDSS];
    __shared__ float Bs[TILE_N * LDSS];

    const int tid  = threadIdx.x;
    const int lane = tid & 31;
    const int wave = tid >> 5;        // 0..7
    const int wm   = wave & 3;        // 4 waves along M -> 32 rows each
    const int wn   = wave >> 2;       // 2 waves along N -> 64 cols each
    const int bm   = blockIdx.x * TILE_M;
    const int bn   = blockIdx.y * TILE_N;

    const int halfsel = lane >> 4;    // 0: K pair {0,1}; 1: K pair {2,3}
    const int lmod    = lane & 15;

    v8f acc[2][4];
#pragma unroll
    for (int i = 0; i < 2; ++i)
#pragma unroll
        for (int j = 0; j < 4; ++j) acc[i][j] = {};

    // Per-thread staging coordinates (4 float4 per tile per thread).
    // Keep the chunk loop rolled: full unroll pushed VGPR count past 256
    // (s_set_vgpr_msb churn + occupancy loss). 64-WMMA inner body is enough ILP.
#pragma unroll 1
    for (int kc = 0; kc < DDIM; kc += KC) {
#pragma unroll
        for (int i = 0; i < 4; ++i) {
            int idx = tid + i * 256;          // 0..1023
            int row = idx >> 3;
            int c4  = (idx & 7) * 4;
            v4f a = *(const v4f*)(xs + (size_t)(bm + row) * DDIM + kc + c4);
            v4f b = *(const v4f*)(ys + (size_t)(bn + row) * DDIM + kc + c4);
            *(v4f*)(As + row * LDSS + c4) = a;
            *(v4f*)(Bs + row * LDSS + c4) = b;
        }
        // Prefetch next chunk while this one is consumed (global_prefetch_b8,
        // no LOADcnt cost).
        if (kc + KC < DDIM) {
#pragma unroll
            for (int i = 0; i < 4; ++i) {
                int idx = tid + i * 256;
                int row = idx >> 3;
                int c4  = (idx & 7) * 4;
                __builtin_prefetch(xs + (size_t)(bm + row) * DDIM + kc + KC + c4, 0, 3);
                __builtin_prefetch(ys + (size_t)(bn + row) * DDIM + kc + KC + c4, 0, 3);
            }
        }
        __syncthreads();

#pragma unroll
        for (int ks = 0; ks < KC; ks += 4) {
            v2f afrag[2], bfrag[4];
#pragma unroll
            for (int mi = 0; mi < 2; ++mi)
                afrag[mi] = *(const v2f*)(As + (wm * 32 + mi * 16 + lmod) * LDSS
                                             + ks + 2 * halfsel);
#pragma unroll
            for (int ni = 0; ni < 4; ++ni)
                bfrag[ni] = *(const v2f*)(Bs + (wn * 64 + ni * 16 + lmod) * LDSS
                                             + ks + 2 * halfsel);
#pragma unroll
            for (int mi = 0; mi < 2; ++mi)
#pragma unroll
                for (int ni = 0; ni < 4; ++ni)
                    acc[mi][ni] = __builtin_amdgcn_wmma_f32_16x16x4_f32(
                        false, afrag[mi], false, bfrag[ni],
                        (short)0, acc[mi][ni], false, false);
        }
        __syncthreads();
    }

    // Fused epilogue. C/D layout: VGPR r, lanes 0-15 -> M=r, lanes 16-31 -> M=8+r; N=lane%16.
    const bool gate = (*cmean) > 0.f;
#pragma unroll
    for (int mi = 0; mi < 2; ++mi) {
#pragma unroll
        for (int ni = 0; ni < 4; ++ni) {
            const int gn  = bn + wn * 64 + ni * 16 + lmod;
            const float yy = y2[gn];
            const float cj = fminf(fmaxf(cutoff[gn], 1e-4f), 0.9999f);
#pragma unroll
            for (int r = 0; r < 8; ++r) {
                const int gm = bm + wm * 32 + mi * 16 + halfsel * 8 + r;
                float xy  = acc[mi][ni][r];
                float dd  = fmaxf(x2[gm] + yy - 2.f * xy, 1e-6f);
                float res = 1.f / (1.f + dd);
                float ci  = fminf(fmaxf(cutoff[gm], 1e-4f), 0.9999f);
                float cmv = 0.5f * (ci + cj);
                float t   = fminf(fmaxf(res - cmv, -1.f), 1.f);   // PHI = 1.0
                float msk = fmaxf(1.f / (1.f + __expf(-t)), 1e-6f);
                out[(size_t)gm * NROWS + gn] = gate ? res * msk : res;
            }
        }
    }
}

// ---------------------------------------------------------------------------
extern "C" void kernel_launch(void* const* d_in, const int* in_sizes, int n_in,
                              void* d_out, int out_size, void* d_ws, size_t ws_size,
                              hipStream_t stream) {
    const float* x      = (const float*)d_in[0];
    const float* y      = (const float*)d_in[1];
    const float* sx     = (const float*)d_in[2];
    const float* sy     = (const float*)d_in[3];
    const float* scale  = (const float*)d_in[4];
    const float* cutoff = (const float*)d_in[5];
    float* out = (float*)d_out;

    float* ws = (float*)d_ws;
    float* xs  = ws;                       // 4096*128
    float* ys  = xs + NROWS * DDIM;        // 4096*128
    float* x2  = ys + NROWS * DDIM;        // 4096
    float* y2  = x2 + NROWS;               // 4096
    float* cm  = y2 + NROWS;               // 1

    scale_rows_kernel<<<NROWS, 128, 0, stream>>>(x, sx, scale, xs, x2);
    scale_rows_kernel<<<NROWS, 128, 0, stream>>>(y, sy, scale, ys, y2);
    mean_kernel<<<1, 256, 0, stream>>>(cutoff, cm, NROWS);

    dim3 grid(NROWS / TILE_M, NROWS / TILE_N);
    cauchy_gemm_kernel<<<grid, 256, 0, stream>>>(xs, ys, x2, y2, cutoff, cm, out);
}